// SSDLoss_81097572483724
// MI455X (gfx1250) — compile-verified
//
#include <hip/hip_runtime.h>
#include <hip/hip_bf16.h>

#define B_ 16
#define A_ 65536
#define C_ 81
#define G_ 32
#define TILE_PER_BLOCK 128            // anchors per 256-thread block (16 per wave)
#define BLOCKS_PER_IMG (A_ / TILE_PER_BLOCK)   // 512

typedef __attribute__((ext_vector_type(2))) float v2f;
typedef __attribute__((ext_vector_type(8))) float v8f;

#define GAS __attribute__((address_space(1)))
#define LAS __attribute__((address_space(3)))

#if defined(__has_builtin)
#if __has_builtin(__builtin_amdgcn_global_load_async_to_lds_b32) && \
    __has_builtin(__builtin_amdgcn_s_wait_asynccnt)
#define SSD_HAVE_ASYNC 1
#endif
#endif
#ifndef SSD_HAVE_ASYNC
#define SSD_HAVE_ASYNC 0
#endif

__device__ __forceinline__ float smooth_l1(float d) {
    d = fabsf(d);
    return (d < 1.0f) ? 0.5f * d * d : d - 0.5f;
}

__global__ __launch_bounds__(256) void ssd_main_kernel(
    const float* __restrict__ logits,   // [B,A,C]
    const float* __restrict__ regs,     // [B,A,4]
    const float* __restrict__ anch,     // [A,4] cxcywh
    const float* __restrict__ gtb,      // [B,G,4] xyxy
    const int*   __restrict__ glab,     // [B,G]
    const int*   __restrict__ gval,     // [B,G]
    float* __restrict__ part)           // [gridDim.x, 4]
{
    __shared__ float sB[G_ * 4];
    __shared__ int   sL[G_];
    __shared__ int   sV[G_];
    __shared__ float wred[8][4];

    const int tid  = threadIdx.x;
    const int b    = blockIdx.x / BLOCKS_PER_IMG;
    const int tile = blockIdx.x % BLOCKS_PER_IMG;

    // ---- Stage GT data for this image into LDS (CDNA5 async path if available)
#if SSD_HAVE_ASYNC
    if (tid < G_ * 4) {
        __builtin_amdgcn_global_load_async_to_lds_b32(
            (GAS int*)(unsigned long long)(gtb + (size_t)b * G_ * 4 + tid),
            (LAS int*)(unsigned int)(unsigned long long)(&sB[tid]), 0, 0);
    } else if (tid < G_ * 4 + G_) {
        int g = tid - G_ * 4;
        __builtin_amdgcn_global_load_async_to_lds_b32(
            (GAS int*)(unsigned long long)(glab + (size_t)b * G_ + g),
            (LAS int*)(unsigned int)(unsigned long long)(&sL[g]), 0, 0);
    } else if (tid < G_ * 4 + 2 * G_) {
        int g = tid - G_ * 4 - G_;
        __builtin_amdgcn_global_load_async_to_lds_b32(
            (GAS int*)(unsigned long long)(gval + (size_t)b * G_ + g),
            (LAS int*)(unsigned int)(unsigned long long)(&sV[g]), 0, 0);
    }
    __builtin_amdgcn_s_wait_asynccnt(0);
#else
    if (tid < G_ * 4)                 sB[tid]            = gtb[(size_t)b * G_ * 4 + tid];
    else if (tid < G_ * 4 + G_)       sL[tid - G_ * 4]   = glab[(size_t)b * G_ + (tid - G_ * 4)];
    else if (tid < G_ * 4 + 2 * G_)   sV[tid - G_ * 5]   = gval[(size_t)b * G_ + (tid - G_ * 5)];
#endif
    __syncthreads();

    const int lane = tid & 31;
    const int half = lane >> 4;       // which class sub-pair this lane covers
    const int ar   = lane & 15;       // anchor row within the wave's 16-anchor tile
    const int wv   = tid >> 5;        // wave id 0..7
    const int a    = tile * TILE_PER_BLOCK + wv * 16 + ar;   // global anchor id
    // NOTE: lanes L and L^16 both handle anchor `ar` (duplicated scalar work);
    // together they feed the 16x4 WMMA A-tile for the sum-exp reduction.

    // ---- Anchor geometry
    const float4 an = ((const float4*)anch)[a];
    const float ax = an.x, ay = an.y, aw = an.z, ah = an.w;
    const float ax1 = ax - 0.5f * aw, ay1 = ay - 0.5f * ah;
    const float ax2 = ax + 0.5f * aw, ay2 = ay + 0.5f * ah;
    const float areaA = fmaxf(ax2 - ax1, 0.f) * fmaxf(ay2 - ay1, 0.f);

    // ---- IoU argmax over G ground-truth boxes (first-max semantics: strict >)
    float best = -2.0f;
    int   bi   = 0;
    for (int g = 0; g < G_; ++g) {
        const float gx1 = sB[g * 4 + 0], gy1 = sB[g * 4 + 1];
        const float gx2 = sB[g * 4 + 2], gy2 = sB[g * 4 + 3];
        const float iw = fmaxf(fminf(ax2, gx2) - fmaxf(ax1, gx1), 0.f);
        const float ih = fmaxf(fminf(ay2, gy2) - fmaxf(ay1, gy1), 0.f);
        const float inter = iw * ih;
        const float areaB = fmaxf(gx2 - gx1, 0.f) * fmaxf(gy2 - gy1, 0.f);
        float iou = inter / (areaA + areaB - inter + 1e-9f);
        iou = (sV[g] != 0) ? iou : -1.0f;
        if (iou > best) { best = iou; bi = g; }
    }
    const bool pos   = best >= 0.5f;
    const bool ign   = (best > 0.4f) && !pos;
    const bool valid = !ign;
    const int  tgt   = pos ? sL[bi] : 0;      // clip(cls_t,0)

    // ---- Smooth-L1 on encoded regression target (masked by pos later)
    const float mx1 = sB[bi * 4 + 0], my1 = sB[bi * 4 + 1];
    const float mx2 = sB[bi * 4 + 2], my2 = sB[bi * 4 + 3];
    const float gcx = 0.5f * (mx1 + mx2), gcy = 0.5f * (my1 + my2);
    const float gw = fmaxf(mx2 - mx1, 1e-6f), gh = fmaxf(my2 - my1, 1e-6f);
    const float t0 = (gcx - ax) / (aw * 0.1f);
    const float t1 = (gcy - ay) / (ah * 0.1f);
    const float t2 = __logf(gw / aw) * 5.0f;   // /0.2
    const float t3 = __logf(gh / ah) * 5.0f;
    const float4 rg = ((const float4*)regs)[(size_t)b * A_ + a];
    const float sl = smooth_l1(rg.x - t0) + smooth_l1(rg.y - t1)
                   + smooth_l1(rg.z - t2) + smooth_l1(rg.w - t3);

    // ---- Cross-entropy: log-softmax over 81 classes, Σexp via v_wmma_f32_16x16x4_f32
    const float* row = logits + ((size_t)b * A_ + (size_t)a) * C_;
    __builtin_prefetch(row + half * 40, 0, 3);   // global_prefetch_b8

    const int cbase = half * 2;                  // this lane covers classes c+cbase, c+cbase+1
    // pass 1: running max over this lane's class subset (c mod 4 in {2h,2h+1})
    float m = -3.4e38f;
    for (int c = 0; c < 80; c += 4) {
        const float x0 = row[c + cbase];
        const float x1 = row[c + cbase + 1];
        m = fmaxf(m, fmaxf(x0, x1));
    }
    m = fmaxf(m, __shfl_xor(m, 16, 32));         // combine the two half-wave subsets
    const float l80 = row[80];
    m = fmaxf(m, l80);

    // pass 2: A-tile = exp(logit - max) (16 anchors x 4 classes), B = ones, C += A*B
    v8f dacc = {0.f, 0.f, 0.f, 0.f, 0.f, 0.f, 0.f, 0.f};
    v2f ones; ones.x = 1.0f; ones.y = 1.0f;
    for (int c = 0; c < 80; c += 4) {
        v2f av;
        av.x = __expf(row[c + cbase]     - m);   // K = 0 (lower) / 2 (upper)
        av.y = __expf(row[c + cbase + 1] - m);   // K = 1 (lower) / 3 (upper)
        dacc = __builtin_amdgcn_wmma_f32_16x16x4_f32(
            false, av, false, ones, (short)0, dacc, false, false);
    }
    // D[m][n] = Σ_k exp  (identical for every column n).
    // Lanes 0-15 hold rows 0-7 in VGPRs 0-7; lanes 16-31 hold rows 8-15.
    // Extract row (ar) with a constant-index select tree + half-wave swap.
    const int i0 = ar & 1, i1 = (ar >> 1) & 1, i2 = (ar >> 2) & 1;
    const float s0 = i0 ? dacc[1] : dacc[0];
    const float s1 = i0 ? dacc[3] : dacc[2];
    const float s2 = i0 ? dacc[5] : dacc[4];
    const float s3 = i0 ? dacc[7] : dacc[6];
    const float u0 = i1 ? s1 : s0;
    const float u1 = i1 ? s3 : s2;
    const float z  = i2 ? u1 : u0;
    const float zp = __shfl_xor(z, 16, 32);
    const bool own = ((((lane >> 3) ^ (lane >> 4)) & 1) == 0);
    float sumexp = own ? z : zp;
    sumexp += __expf(l80 - m);                   // class 80 remainder (81 = 20*4 + 1)

    const float ce = __logf(sumexp) + m - row[tgt];

    // ---- Per-lane contributions (only lower half counts; upper half is duplicate)
    const float wsel = (lane < 16) ? 1.0f : 0.0f;
    float c0 = wsel * (valid ? ce  : 0.f);
    float c1 = wsel * (valid ? 1.f : 0.f);
    float c2 = wsel * (pos   ? sl  : 0.f);
    float c3 = wsel * (pos   ? 1.f : 0.f);
    for (int off = 16; off >= 1; off >>= 1) {
        c0 += __shfl_xor(c0, off, 32);
        c1 += __shfl_xor(c1, off, 32);
        c2 += __shfl_xor(c2, off, 32);
        c3 += __shfl_xor(c3, off, 32);
    }
    if (lane == 0) {
        wred[wv][0] = c0; wred[wv][1] = c1; wred[wv][2] = c2; wred[wv][3] = c3;
    }
    __syncthreads();
    if (tid == 0) {
        float p0 = 0.f, p1 = 0.f, p2 = 0.f, p3 = 0.f;
        for (int w = 0; w < 8; ++w) {            // fixed order -> deterministic
            p0 += wred[w][0]; p1 += wred[w][1]; p2 += wred[w][2]; p3 += wred[w][3];
        }
        float* o = part + (size_t)blockIdx.x * 4;
        o[0] = p0; o[1] = p1; o[2] = p2; o[3] = p3;
    }
}

__global__ __launch_bounds__(64) void ssd_finalize_kernel(
    const float* __restrict__ part, float* __restrict__ out)
{
    __shared__ float s[B_ * 4];
    const int t = threadIdx.x;                   // 0..63 -> (image, quantity)
    const int b = t >> 2, q = t & 3;
    float sum = 0.f;
    for (int i = 0; i < BLOCKS_PER_IMG; ++i)     // fixed order -> deterministic
        sum += part[((size_t)b * BLOCKS_PER_IMG + i) * 4 + q];
    s[t] = sum;
    __syncthreads();
    if (t == 0) {
        float tot = 0.f;
        for (int bb = 0; bb < B_; ++bb) {
            const float ce = s[bb * 4 + 0], nv = s[bb * 4 + 1];
            const float sl = s[bb * 4 + 2], np = s[bb * 4 + 3];
            if (nv > 0.f) tot += ce / fmaxf(nv, 1.f);
            if (np > 0.f) tot += sl / fmaxf(4.f * np, 1.f);
        }
        out[0] = tot;
    }
}

extern "C" void kernel_launch(void* const* d_in, const int* in_sizes, int n_in,
                              void* d_out, int out_size, void* d_ws, size_t ws_size,
                              hipStream_t stream) {
    const float* logits = (const float*)d_in[0];   // [B,A,C] f32
    const float* regs   = (const float*)d_in[1];   // [B,A,4] f32
    const float* anch   = (const float*)d_in[2];   // [A,4]   f32
    const float* gtb    = (const float*)d_in[3];   // [B,G,4] f32
    const int*   glab   = (const int*)d_in[4];     // [B,G]   int
    const int*   gval   = (const int*)d_in[5];     // [B,G]   int (bool)
    float* part = (float*)d_ws;                    // [B*BLOCKS_PER_IMG, 4] partials

    const int nblocks = B_ * BLOCKS_PER_IMG;       // 8192
    ssd_main_kernel<<<nblocks, 256, 0, stream>>>(logits, regs, anch, gtb, glab, gval, part);
    ssd_finalize_kernel<<<1, 64, 0, stream>>>(part, (float*)d_out);
}